// SSIM_66099546686050
// MI455X (gfx1250) — compile-verified
//
#include <hip/hip_runtime.h>

typedef __attribute__((ext_vector_type(2))) float v2f;
typedef __attribute__((ext_vector_type(8))) float v8f;

#define IMG_H 256
#define IMG_W 256
#define VALID 250          // 256 - (7-1)
#define NIMG  384          // 128 * 3 channel-images
#define TILE_R 32          // output rows per workgroup
#define TILE_C 64          // output cols per workgroup
#define REG_R 38           // TILE_R + 6 input rows needed
#define REG_C 70           // TILE_C + 6 input cols needed
#define LDSP  72           // padded LDS pitch for input tiles (async chunks fill 72)
#define CW    25           // padded width of per-wave C spill (coprime w/ 64 banks)
#define NTR   8            // ceil(250/32)
#define NTC   4            // ceil(250/64)
#define NWG   (NIMG * NTR * NTC)   // 12288 partials

#if defined(__gfx1250__) && __has_builtin(__builtin_amdgcn_global_load_async_to_lds_b128) && __has_builtin(__builtin_amdgcn_s_wait_asynccnt)
#define USE_ASYNC_LDS 1
#else
#define USE_ASYNC_LDS 0
#endif

#if USE_ASYNC_LDS
typedef __attribute__((__vector_size__(16))) int v4i;
typedef __attribute__((address_space(1))) v4i gv4i;   // global (__device__) int4
typedef __attribute__((address_space(3))) v4i lv4i;   // LDS (__shared__) int4

__device__ __forceinline__ gv4i* as_global_v4(const void* p) {
  return (gv4i*)(unsigned long long)p;                 // 64-bit global address
}
__device__ __forceinline__ lv4i* as_lds_v4(void* p) {
  return (lv4i*)(unsigned int)(unsigned long long)p;   // low 32 bits = LDS offset
}
#endif

__device__ __forceinline__ float band01(int m, int k) {
  // 7-wide band of ones: 1 iff m <= k <= m+6
  return (k >= m && k <= m + 6) ? 1.0f : 0.0f;
}

__global__ __launch_bounds__(256)
void ssim_tile_kernel(const float* __restrict__ X, const float* __restrict__ Y,
                      float* __restrict__ partial) {
  __shared__ float sX[REG_R][LDSP];
  __shared__ float sY[REG_R][LDSP];
  __shared__ float sC[8][5][16][CW];   // per-wave vertical-pass spill, all 5 quantities
  __shared__ float sPart[8];

  const int tid  = threadIdx.x;
  const int lane = tid & 31;
  const int wave = tid >> 5;
  const int hi   = lane >> 4;   // half-wave select
  const int idx  = lane & 15;

  const int tj  = blockIdx.x;   // 0..3
  const int ti  = blockIdx.y;   // 0..7
  const int img = blockIdx.z;   // 0..383
  const int r0  = ti * TILE_R;
  const int c0  = tj * TILE_C;

  const float* xb = X + (size_t)img * (IMG_H * IMG_W);
  const float* yb = Y + (size_t)img * (IMG_H * IMG_W);

#if USE_ASYNC_LDS
  // ---- async-engine fill: 18 x B128 chunks per row (72 cols), 38 rows, 2 arrays ----
  // Chunk global start is clamped inside the image; clamped chunks only feed
  // masked (invalid) outputs and stay finite.
  for (int e = tid; e < 2 * REG_R * 18; e += 256) {
    int a  = (e >= REG_R * 18) ? 1 : 0;
    int e2 = e - a * (REG_R * 18);
    int lr = e2 / 18;
    int ck = e2 - lr * 18;
    int gr = min(r0 + lr, IMG_H - 1);
    int gc = min(c0 + 4 * ck, IMG_W - 4);
    const float* src = (a ? yb : xb) + gr * IMG_W + gc;
    float* dst = a ? &sY[lr][4 * ck] : &sX[lr][4 * ck];
    __builtin_amdgcn_global_load_async_to_lds_b128(as_global_v4(src), as_lds_v4(dst), 0, 0);
  }
  __builtin_amdgcn_s_wait_asynccnt(0);
  __syncthreads();
#else
  // ---- fallback: cooperative, clamped scalar load of the 38x70 region ----
  for (int e = tid; e < REG_R * REG_C; e += 256) {
    int lr = e / REG_C;
    int lc = e - lr * REG_C;
    int gr = min(r0 + lr, IMG_H - 1);
    int gc = min(c0 + lc, IMG_W - 1);
    sX[lr][lc] = xb[gr * IMG_W + gc];
    sY[lr][lc] = yb[gr * IMG_W + gc];
  }
  __syncthreads();
#endif

  const int wr = (wave >> 2) << 4;      // wave's row origin inside tile (0/16)
  const int wc = (wave & 3) << 4;       // wave's col origin inside tile (0..48)

  // ---- band fragments, computed ONCE: serve as A (vertical) and B (horizontal) ----
  v2f wb[6];
#pragma unroll
  for (int kk = 0; kk < 6; ++kk)
#pragma unroll
    for (int v = 0; v < 2; ++v)
      wb[kk][v] = band01(idx, 4 * kk + v + 2 * hi);

  // ---- vertical 7-tap pass: C_q(16x22) = Wband(16x24) * Q_q(24x22), all 5 q fused ----
#pragma unroll
  for (int t = 0; t < 2; ++t) {            // column sub-tiles 0..15, 16..21
    v8f a0 = {0.f,0.f,0.f,0.f,0.f,0.f,0.f,0.f};
    v8f a1 = a0, a2 = a0, a3 = a0, a4 = a0;
#pragma unroll
    for (int kk = 0; kk < 6; ++kk) {       // K = 24 in chunks of 4
      v2f bx, by, bxx, byy, bxy;
#pragma unroll
      for (int v = 0; v < 2; ++v) {
        int kg = 4 * kk + v + 2 * hi;      // global K index of this element
        int rr = wr + min(kg, 21);         // clamp: band weight is 0 beyond 21
        int cc = wc + min(16 * t + idx, 21);
        float xv = sX[rr][cc];
        float yv = sY[rr][cc];
        bx[v]  = xv;      by[v]  = yv;
        bxx[v] = xv * xv; byy[v] = yv * yv; bxy[v] = xv * yv;
      }
      a0 = __builtin_amdgcn_wmma_f32_16x16x4_f32(false, wb[kk], false, bx,  (short)0, a0, false, false);
      a1 = __builtin_amdgcn_wmma_f32_16x16x4_f32(false, wb[kk], false, by,  (short)0, a1, false, false);
      a2 = __builtin_amdgcn_wmma_f32_16x16x4_f32(false, wb[kk], false, bxx, (short)0, a2, false, false);
      a3 = __builtin_amdgcn_wmma_f32_16x16x4_f32(false, wb[kk], false, byy, (short)0, a3, false, false);
      a4 = __builtin_amdgcn_wmma_f32_16x16x4_f32(false, wb[kk], false, bxy, (short)0, a4, false, false);
    }
    // spill D (M=v+8*hi, N=idx) to per-wave LDS, columns t*16 + idx (< 24)
    if (t == 0 || idx < 8) {
#pragma unroll
      for (int v = 0; v < 8; ++v) {
        int m = v + 8 * hi;
        int n = 16 * t + idx;
        sC[wave][0][m][n] = a0[v];
        sC[wave][1][m][n] = a1[v];
        sC[wave][2][m][n] = a2[v];
        sC[wave][3][m][n] = a3[v];
        sC[wave][4][m][n] = a4[v];
      }
    }
  }

  // ---- horizontal 7-tap pass: S_q(16x16) = C_q(16x24) * Wt(24x16) ----
  v8f S[5];
#pragma unroll
  for (int q = 0; q < 5; ++q) {
    v8f acc = {0.f,0.f,0.f,0.f,0.f,0.f,0.f,0.f};
#pragma unroll
    for (int kk = 0; kk < 6; ++kk) {
      v2f aC;
#pragma unroll
      for (int v = 0; v < 2; ++v)
        aC[v] = sC[wave][q][idx][4 * kk + v + 2 * hi];   // A: M=idx, K=kg
      acc = __builtin_amdgcn_wmma_f32_16x16x4_f32(false, aC, false, wb[kk],
                                                  (short)0, acc, false, false);
    }
    S[q] = acc;
  }

  // ---- SSIM map + masked reduction ----
  const float invNP = 1.0f / 49.0f;
  const float covn  = 49.0f / 48.0f;         // sample-covariance normalization
  const float C1f   = 0.01f * 0.01f;         // (K1*data_range)^2
  const float C2f   = 0.03f * 0.03f;         // (K2*data_range)^2

  float lsum = 0.0f;
#pragma unroll
  for (int v = 0; v < 8; ++v) {
    int gi = r0 + wr + v + 8 * hi;           // D layout: M = v + 8*hi
    int gj = c0 + wc + idx;                  //           N = idx
    float ux  = S[0][v] * invNP;
    float uy  = S[1][v] * invNP;
    float uxx = S[2][v] * invNP;
    float uyy = S[3][v] * invNP;
    float uxy = S[4][v] * invNP;
    float vx  = covn * (uxx - ux * ux);
    float vy  = covn * (uyy - uy * uy);
    float vxy = covn * (uxy - ux * uy);
    float num = (2.0f * ux * uy + C1f) * (2.0f * vxy + C2f);
    float den = (ux * ux + uy * uy + C1f) * (vx + vy + C2f);
    float sv  = num / den;
    lsum += (gi < VALID && gj < VALID) ? sv : 0.0f;
  }
#pragma unroll
  for (int off = 16; off > 0; off >>= 1) lsum += __shfl_xor(lsum, off, 32);
  if (lane == 0) sPart[wave] = lsum;
  __syncthreads();
  if (tid == 0) {
    float tot = 0.0f;
#pragma unroll
    for (int w = 0; w < 8; ++w) tot += sPart[w];
    partial[(img * NTR + ti) * NTC + tj] = tot;
  }
}

__global__ __launch_bounds__(256)
void ssim_finalize_kernel(const float* __restrict__ partial,
                          float* __restrict__ out) {
  __shared__ float sred[256];
  float s = 0.0f;
  for (int i = threadIdx.x; i < NWG; i += 256) s += partial[i];  // fixed order
  sred[threadIdx.x] = s;
  __syncthreads();
  for (int st = 128; st > 0; st >>= 1) {
    if ((int)threadIdx.x < st) sred[threadIdx.x] += sred[threadIdx.x + st];
    __syncthreads();
  }
  if (threadIdx.x == 0)
    out[0] = sred[0] * (1.0f / 24000000.0f);   // mean over 384*250*250 windows
}

extern "C" void kernel_launch(void* const* d_in, const int* in_sizes, int n_in,
                              void* d_out, int out_size, void* d_ws, size_t ws_size,
                              hipStream_t stream) {
  const float* X = (const float*)d_in[0];   // predicted
  const float* Y = (const float*)d_in[1];   // target
  // d_in[2] = win_size (scalar 7): kernel is specialized for 7.
  float* partial = (float*)d_ws;            // NWG * 4 = 48 KB scratch

  dim3 grid(NTC, NTR, NIMG);
  ssim_tile_kernel<<<grid, 256, 0, stream>>>(X, Y, partial);
  ssim_finalize_kernel<<<1, 256, 0, stream>>>(partial, (float*)d_out);
}